// CLRBP_23124103922240
// MI455X (gfx1250) — compile-verified
//
#include <hip/hip_runtime.h>

#define D_DIM 256
#define KTOT  65536          // D_DIM * D_DIM
#define BATCH 1024
#define NC    16             // channels
#define KW    512            // K elements per wave
#define KCHUNK 64            // K per prefetch chunk (16 WMMA ops)
#define WAVES 8              // waves per workgroup (256 threads, wave32)

typedef float v2f __attribute__((ext_vector_type(2)));
typedef float v8f __attribute__((ext_vector_type(8)));

// ---------------------------------------------------------------------------
// Kernel 1: materialize W^T in pair-interleaved layout + zero the scores acc.
//   W[c,k] = g*A1[c, i%64, j%64] + (1-g)*A2[c,i,j],  k = i*256 + j
//   A1 = u1 u1^T - v1 v1^T (rank 4+4 over 64),  A2 = u2 u2^T - v2 v2^T (rank 1+1)
//   wt2 layout: wt2[(k>>1)*32 + c*2 + (k&1)]  -> per-lane contiguous b64 B-frags
// ---------------------------------------------------------------------------
__global__ void build_w_kernel(const float* __restrict__ w1,
                               const float* __restrict__ w2,
                               const float* __restrict__ l,
                               float* __restrict__ wt2,
                               float* __restrict__ scores) {
  const int tid = blockIdx.x * blockDim.x + threadIdx.x;   // 0 .. 1M-1
  if (tid < BATCH * NC) scores[tid] = 0.0f;                // zero accumulator

  const int c = tid & 15;
  const int k = tid >> 4;            // 0..65535
  const int i = k >> 8;              // row in d x d
  const int j = k & 255;             // col in d x d
  const int i64 = i & 63, j64 = j & 63;

  const float* w1c = w1 + c * 64 * 8;          // (64, 8): [:, :4]=v1, [:, 4:]=u1
  float a1 = 0.0f;
#pragma unroll
  for (int r = 0; r < 4; ++r) {
    const float ui = w1c[i64 * 8 + 4 + r], uj = w1c[j64 * 8 + 4 + r];
    const float vi = w1c[i64 * 8 + r],     vj = w1c[j64 * 8 + r];
    a1 += ui * uj - vi * vj;
  }

  const float* w2c = w2 + c * 256 * 2;         // (256, 2): [:,0]=v2, [:,1]=u2
  const float u2i = w2c[i * 2 + 1], u2j = w2c[j * 2 + 1];
  const float v2i = w2c[i * 2 + 0], v2j = w2c[j * 2 + 0];
  const float a2 = u2i * u2j - v2i * v2j;

  const float g = 1.0f / (1.0f + __expf(-l[0]));
  const float w = g * a1 + (1.0f - g) * a2;

  wt2[(size_t)(k >> 1) * 32 + c * 2 + (k & 1)] = w;
}

// ---------------------------------------------------------------------------
// Kernel 2: scores += x @ W^T via V_WMMA_F32_16X16X4_F32.
//   grid = (KTOT/(KW*WAVES), BATCH/16); 256 threads = 8 waves.
//   Each wave: 16 sample-rows x 16 channels, K-slice of KW, 128 WMMA ops.
//   A-frag (16x4 f32): lanes 0-15 hold K=0,1 ; lanes 16-31 hold K=2,3.
//   B-frag: one b64 from pair-interleaved wt2; both streams are linear in kk,
//   so bases are hoisted and the unrolled body uses immediate offsets only.
//   Partials reduced via LDS (ds_add_f32), then global_atomic_add_f32.
// ---------------------------------------------------------------------------
__global__ void gemm_wmma_kernel(const float* __restrict__ x,
                                 const float* __restrict__ wt2,
                                 float* __restrict__ scores) {
  __shared__ float tile[NC * NC];

  const int lane    = threadIdx.x & 31;
  const int wave    = threadIdx.x >> 5;
  const int mtile   = blockIdx.y;                       // 0..63
  const int kbase   = (blockIdx.x * WAVES + wave) * KW; // wave's K slice
  const int halfsel = lane >> 4;                        // 0 | 1
  const int khalf   = halfsel * 2;                      // K sub-offset 0 | 2
  const int nlane   = lane & 15;                        // M index (A) / N index (B,D)

  // Per-lane linear base pointers; both advance linearly with kk.
  const float* aptr = x + (size_t)(mtile * NC + nlane) * KTOT + kbase + khalf;
  const float* bptr = wt2 + ((size_t)(kbase + khalf) >> 1) * 32 + nlane * 2;

  if (threadIdx.x < NC * NC) tile[threadIdx.x] = 0.0f;
  __syncthreads();

  v8f acc = {};
  for (int kc = 0; kc < KW; kc += KCHUNK) {
    // one stream-ahead prefetch per chunk, outside the unrolled body
    __builtin_prefetch(aptr + kc + 2 * KCHUNK, 0, 1);   // global_prefetch_b8
#pragma unroll
    for (int kk = 0; kk < KCHUNK; kk += 4) {
      const int k = kc + kk;
      v2f a = *(const v2f*)(aptr + k);                  // A 16x4: imm offsets
      v2f b = *(const v2f*)(bptr + (size_t)k * 16);     // B 4x16: imm offsets
      acc = __builtin_amdgcn_wmma_f32_16x16x4_f32(
          /*neg_a=*/false, a, /*neg_b=*/false, b,
          /*c_mod=*/(short)0, acc, /*reuse_a=*/false, /*reuse_b=*/false);
    }
  }

  // D layout: lane l, vgpr j -> (M = j + 8*(l>=16), N = l%16)
#pragma unroll
  for (int j = 0; j < 8; ++j) {
    const int M = j + halfsel * 8;
    atomicAdd(&tile[M * NC + nlane], acc[j]);           // ds_add_f32
  }
  __syncthreads();

  if (threadIdx.x < NC * NC)
    unsafeAtomicAdd(&scores[mtile * NC * NC + threadIdx.x], tile[threadIdx.x]);
}

// ---------------------------------------------------------------------------
// Kernel 3: out = softmax(scores + b) over the 16 channels, 1 thread per row.
// ---------------------------------------------------------------------------
__global__ void softmax_kernel(const float* __restrict__ scores,
                               const float* __restrict__ bvec,
                               float* __restrict__ out) {
  const int row = blockIdx.x * blockDim.x + threadIdx.x;
  if (row >= BATCH) return;

  float s[NC];
  float mx = -3.402823466e38f;
#pragma unroll
  for (int c = 0; c < NC; ++c) {
    s[c] = scores[row * NC + c] + bvec[c];
    mx = fmaxf(mx, s[c]);
  }
  float sum = 0.0f;
#pragma unroll
  for (int c = 0; c < NC; ++c) {
    s[c] = __expf(s[c] - mx);
    sum += s[c];
  }
  const float inv = 1.0f / sum;
#pragma unroll
  for (int c = 0; c < NC; ++c) out[row * NC + c] = s[c] * inv;
}

// ---------------------------------------------------------------------------
extern "C" void kernel_launch(void* const* d_in, const int* in_sizes, int n_in,
                              void* d_out, int out_size, void* d_ws, size_t ws_size,
                              hipStream_t stream) {
  const float* x  = (const float*)d_in[0];  // (1024, 256, 256)
  const float* w1 = (const float*)d_in[1];  // (16, 64, 8)
  const float* w2 = (const float*)d_in[2];  // (16, 256, 2)
  const float* l  = (const float*)d_in[3];  // (16,)
  const float* b  = (const float*)d_in[4];  // (16,)

  float* scores = (float*)d_ws;              // 1024*16 f32 = 64 KB
  float* wt2    = (float*)d_ws + BATCH * NC; // 65536*16 f32 = 4 MB

  // 1) build W^T (pair-interleaved) + zero scores: 1M threads
  build_w_kernel<<<(KTOT * NC) / 256, 256, 0, stream>>>(w1, w2, l, wt2, scores);

  // 2) WMMA GEMM with K-split accumulation
  dim3 grid(KTOT / (KW * WAVES), BATCH / NC);  // (16, 64)
  gemm_wmma_kernel<<<grid, 256, 0, stream>>>(x, wt2, scores);

  // 3) bias + softmax
  softmax_kernel<<<BATCH / 256, 256, 0, stream>>>(scores, b, (float*)d_out);
}